// AlignmentEncoder_6975026888885
// MI455X (gfx1250) — compile-verified
//
#include <hip/hip_runtime.h>
#include <hip/hip_bf16.h>
#include <cstdint>

// ---------------------------------------------------------------------------
// AlignmentEncoder forward for MI455X (gfx1250, wave32).
// All GEMM-shaped work (conv3, conv1, qk einsum) uses V_WMMA_F32_16X16X4_F32.
// fp32 throughout: the op is HBM-bound (~0.55 GB moved vs ~72 GFLOP).
// k=1 GEMMs use double-buffered GLOBAL_LOAD_ASYNC_TO_LDS (ASYNCcnt) staging;
// conv3 keeps sync staging (padding + speaker add are fused into the staged
// values, which a raw async copy can't do).
// ---------------------------------------------------------------------------

typedef float v2f __attribute__((ext_vector_type(2)));
typedef float v8f __attribute__((ext_vector_type(8)));

#define WMMA_F32(a, b, c) \
    __builtin_amdgcn_wmma_f32_16x16x4_f32(false, (a), false, (b), (short)0, (c), false, false)

#define NB   32
#define NT1  2048
#define NT2  512
#define NC   80      // n_att channels
#define TEMP 0.0005f
#define CK   16      // ci chunk staged in LDS per iteration

// --------------------------- CDNA5 async-to-LDS ----------------------------
#if __has_builtin(__builtin_amdgcn_global_load_async_to_lds_b32)
#  define HAVE_ASYNC_LDS 1
#else
#  define HAVE_ASYNC_LDS 0
#endif

#if __has_builtin(__builtin_amdgcn_s_wait_asynccnt)
#  define WAIT_ASYNC0() __builtin_amdgcn_s_wait_asynccnt(0)
#else
#  define WAIT_ASYNC0() asm volatile("s_wait_asynccnt 0" ::: "memory")
#endif

#if HAVE_ASYNC_LDS
// Builtin signature (from clang diagnostic): arg0 = AS1 ("__device__") int*,
// arg1 = AS3 ("__shared__") int*, then imm offset, imm cpol.
// Generic global pointers equal their AS1 value; generic LDS pointers carry
// the AS3 offset in the low 32 bits -> integer casts avoid addrspacecast.
typedef __attribute__((address_space(1))) int g_int;
typedef __attribute__((address_space(3))) int lds_int;
#define GPTR(p) ((g_int*)(unsigned long long)(const void*)(p))
#define LPTR(p) ((lds_int*)(unsigned int)(unsigned long long)(const void*)(p))
#define ASYNC_CP_B32(gsrc, ldst) \
    __builtin_amdgcn_global_load_async_to_lds_b32(GPTR(gsrc), LPTR(ldst), 0, 0)
#endif

// ---------------------------------------------------------------------------
// Speaker projections: k_spk[b,c] = emb[b]·kspk_w[c] + kb ; same for q_spk.
// ---------------------------------------------------------------------------
__global__ void spk_proj(const float* __restrict__ emb,
                         const float* __restrict__ kw, const float* __restrict__ kb,
                         const float* __restrict__ qw, const float* __restrict__ qb,
                         float* __restrict__ kspk, float* __restrict__ qspk)
{
    const int b = blockIdx.x;
    const int c = threadIdx.x;
    const float* e = emb + (size_t)b * 512;
    if (c < 512) {
        const float* wr = kw + (size_t)c * 512;
        float s = kb[c];
        for (int i = 0; i < 512; ++i) s += e[i] * wr[i];
        kspk[b * 512 + c] = s;
    } else if (c < 512 + NC) {
        const int cq = c - 512;
        const float* wr = qw + (size_t)cq * 512;
        float s = qb[cq];
        for (int i = 0; i < 512; ++i) s += e[i] * wr[i];
        qspk[b * NC + cq] = s;
    }
}

// ---------------------------------------------------------------------------
// k=3 'same' conv as WMMA GEMM. Block tile: 16 co x 128 t (8 waves x 16 t).
// Sync LDS staging (zero padding + speaker add are fused into staged values).
// f32 WMMA fragment layout: A(m,k): lane = m + 16*(k>=2), vgpr = k&1.
//                           B(k,n): lane = n + 16*(k>=2), vgpr = k&1.
// ---------------------------------------------------------------------------
__launch_bounds__(256)
__global__ void conv3_relu_wmma(const float* __restrict__ x,    // [B,Cin,T]
                                const float* __restrict__ spk,  // [B,Cin]
                                const float* __restrict__ w,    // [Cout,Cin,3]
                                const float* __restrict__ bias, // [Cout]
                                float* __restrict__ y,          // [B,Cout,T]
                                int Cin, int Cout, int T)
{
    __shared__ float Xs[CK][132];       // cols = t in [tblk-1, tblk+129)
    __shared__ float Ws[16 * CK * 3];   // [co][ci][k] flat

    const int tid  = threadIdx.x;
    const int lane = tid & 31;
    const int wv   = tid >> 5;
    const int h    = lane >> 4;
    const int r    = lane & 15;
    const int tblk = blockIdx.x * 128;
    const int co0  = blockIdx.y * 16;
    const int b    = blockIdx.z;

    const float* xb = x   + (size_t)b * Cin * T;
    const float* sb = spk + (size_t)b * Cin;

    v8f acc = {};
    for (int cc = 0; cc < Cin; cc += CK) {
        __syncthreads();
        // stage weights: 16 co x CK ci x 3 taps (contiguous 48-elem runs per co)
        for (int e = tid; e < 16 * CK * 3; e += 256) {
            const int co = e / (CK * 3);
            const int j  = e % (CK * 3);
            Ws[e] = w[(size_t)(co0 + co) * Cin * 3 + (size_t)cc * 3 + j];
        }
        // stage input: CK ci x 130 cols, zero-padded, + speaker bias
        for (int e = tid; e < CK * 130; e += 256) {
            const int ci  = e / 130;
            const int col = e % 130;
            const int t   = tblk - 1 + col;
            float v = 0.0f;
            if ((unsigned)t < (unsigned)T)
                v = xb[(size_t)(cc + ci) * T + t] + sb[cc + ci];
            Xs[ci][col] = v;
            if (cc + CK < Cin && (unsigned)t < (unsigned)T)
                __builtin_prefetch(&xb[(size_t)(cc + CK + ci) * T + t], 0, 0);
        }
        __syncthreads();
        // branch-free WMMA loop: wave wv owns t cols [tblk+wv*16, +16)
        #pragma unroll
        for (int k = 0; k < 3; ++k) {
            #pragma unroll
            for (int c4 = 0; c4 < CK; c4 += 4) {
                const int c = c4 + 2 * h;
                v2f a, bb;
                a.x  = Ws[r * (CK * 3) + c * 3 + k];
                a.y  = Ws[r * (CK * 3) + (c + 1) * 3 + k];
                bb.x = Xs[c][wv * 16 + r + k];
                bb.y = Xs[c + 1][wv * 16 + r + k];
                acc = WMMA_F32(a, bb, acc);
            }
        }
    }
    // C layout: row m = v + 8*h, col n = r
    for (int v = 0; v < 8; ++v) {
        const int co = co0 + v + 8 * h;
        float val = acc[v] + bias[co];
        val = fmaxf(val, 0.0f);
        y[((size_t)b * Cout + co) * T + (tblk + wv * 16 + r)] = val;
    }
}

// ---------------------------------------------------------------------------
// k=1 conv == per-batch GEMM [Cout x T] = W[Cout x Cin] * X[Cin x T].
// Block tile 16 co x 128 t. Double-buffered async global->LDS staging:
//   iter i: s_wait_asynccnt 0 ; barrier ; issue chunk i+1 ; compute chunk i.
// The barrier after the wait makes everyone's chunk-i data visible AND
// guarantees no wave still reads the buffer chunk i+1 overwrites.
// ---------------------------------------------------------------------------
__launch_bounds__(256)
__global__ void gemm_conv1_wmma(const float* __restrict__ x,    // [B,Cin,T]
                                const float* __restrict__ w,    // [Cout,Cin]
                                const float* __restrict__ bias, // [Cout]
                                float* __restrict__ y,          // [B,Cout,T]
                                int Cin, int Cout, int T, int relu)
{
    const int tid  = threadIdx.x;
    const int lane = tid & 31;
    const int wv   = tid >> 5;
    const int h    = lane >> 4;
    const int r    = lane & 15;
    const int tblk = blockIdx.x * 128;
    const int co0  = blockIdx.y * 16;
    const int b    = blockIdx.z;

    const float* xb = x + (size_t)b * Cin * T;

#if HAVE_ASYNC_LDS
    __shared__ float Xs[2][CK][130];
    __shared__ float Ws[2][16][CK];

    // issue one chunk's copies: 1 (W) + 8 (X) async-load instructions per wave
    auto issue = [&](int p, int cc) {
        {
            const int co = tid / CK, j = tid % CK;   // 256 threads == 16*CK
            ASYNC_CP_B32(&w[(size_t)(co0 + co) * Cin + cc + j], &Ws[p][co][j]);
        }
        #pragma unroll
        for (int i = 0; i < (CK * 128) / 256; ++i) {
            const int ee = tid + i * 256;
            const int ci = ee >> 7, col = ee & 127;
            ASYNC_CP_B32(&xb[(size_t)(cc + ci) * T + tblk + col], &Xs[p][ci][col]);
        }
    };

    v8f acc = {};
    issue(0, 0);
    int p = 0;
    for (int cc = 0; cc < Cin; cc += CK) {
        WAIT_ASYNC0();
        __syncthreads();
        if (cc + CK < Cin) issue(p ^ 1, cc + CK);
        #pragma unroll
        for (int c4 = 0; c4 < CK; c4 += 4) {
            const int c = c4 + 2 * h;
            v2f a, bb;
            a.x  = Ws[p][r][c];
            a.y  = Ws[p][r][c + 1];
            bb.x = Xs[p][c][wv * 16 + r];
            bb.y = Xs[p][c + 1][wv * 16 + r];
            acc = WMMA_F32(a, bb, acc);
        }
        p ^= 1;
    }
#else
    __shared__ float Xs[CK][130];
    __shared__ float Ws[16][CK];

    v8f acc = {};
    for (int cc = 0; cc < Cin; cc += CK) {
        __syncthreads();
        for (int e = tid; e < 16 * CK; e += 256) {
            const int co = e / CK, j = e % CK;
            Ws[co][j] = w[(size_t)(co0 + co) * Cin + cc + j];
        }
        for (int e = tid; e < CK * 128; e += 256) {
            const int ci = e >> 7, col = e & 127;
            Xs[ci][col] = xb[(size_t)(cc + ci) * T + tblk + col];
        }
        __syncthreads();
        #pragma unroll
        for (int c4 = 0; c4 < CK; c4 += 4) {
            const int c = c4 + 2 * h;
            v2f a, bb;
            a.x  = Ws[r][c];
            a.y  = Ws[r][c + 1];
            bb.x = Xs[c][wv * 16 + r];
            bb.y = Xs[c + 1][wv * 16 + r];
            acc = WMMA_F32(a, bb, acc);
        }
    }
#endif
    for (int v = 0; v < 8; ++v) {
        const int co = co0 + v + 8 * h;
        float val = acc[v] + bias[co];
        if (relu) val = fmaxf(val, 0.0f);
        y[((size_t)b * Cout + co) * T + (tblk + wv * 16 + r)] = val;
    }
}

// ---------------------------------------------------------------------------
// q2[b,t] = sum_c x[b,c,t]^2
// ---------------------------------------------------------------------------
__global__ void sqsum(const float* __restrict__ x, float* __restrict__ out,
                      int C, int T)
{
    const int idx = blockIdx.x * blockDim.x + threadIdx.x;
    const int b = idx / T, t = idx % T;
    const float* xb = x + (size_t)b * C * T + t;
    float s = 0.0f;
    for (int c = 0; c < C; ++c) { const float v = xb[(size_t)c * T]; s += v * v; }
    out[idx] = s;
}

// ---------------------------------------------------------------------------
// Fused attention tail. Block = 16 query rows x full 512 key positions.
// Phase 1: qk via WMMA; the 80x16 query tile (identical for all 8 waves) is
//          staged once in LDS (async copy); ke fragments come from
//          (L2-resident) global. scores = -temp*(q2+k2-2qk) in LDS.
// Phase 2: per-row log-softmax, +log(prior+1e-8) -> attn_logprob;
//          mask -> -1e9, second softmax -> attn. One write per output element.
// ---------------------------------------------------------------------------
__launch_bounds__(256)
__global__ void attn_fused(const float* __restrict__ qe,    // [B,80,2048]
                           const float* __restrict__ ke,    // [B,80,512]
                           const float* __restrict__ q2,    // [B,2048]
                           const float* __restrict__ k2,    // [B,512]
                           const float* __restrict__ prior, // [B,2048,512]
                           const unsigned char* __restrict__ mask, // [B,512] (JAX bool = 1B)
                           float* __restrict__ attn_out,
                           float* __restrict__ logp_out)
{
    __shared__ float sc[16][NT2 + 4];
    __shared__ float Qs[NC][16];
    __shared__ float red[16][16];

    const int tid  = threadIdx.x;
    const int lane = tid & 31;
    const int wv   = tid >> 5;
    const int h    = lane >> 4;
    const int r    = lane & 15;
    const int t10  = blockIdx.x * 16;
    const int b    = blockIdx.y;

    const float* qeb = qe + (size_t)b * NC * NT1;
    const float* keb = ke + (size_t)b * NC * NT2;

    // stage 80x16 query tile once for the whole block
#if HAVE_ASYNC_LDS
    #pragma unroll
    for (int i = 0; i < (NC * 16) / 256; ++i) {
        const int e = tid + i * 256;
        const int c = e >> 4, tt = e & 15;
        ASYNC_CP_B32(&qeb[(size_t)c * NT1 + t10 + tt], &Qs[c][tt]);
    }
    WAIT_ASYNC0();
#else
    for (int e = tid; e < NC * 16; e += 256) {
        const int c = e >> 4, tt = e & 15;
        Qs[c][tt] = qeb[(size_t)c * NT1 + t10 + tt];
    }
#endif
    __syncthreads();

    v8f acc[4] = {};
    for (int c4 = 0; c4 < NC; c4 += 4) {
        const int c = c4 + 2 * h;
        v2f a;
        a.x = Qs[c][r];
        a.y = Qs[c + 1][r];
        #pragma unroll
        for (int tt = 0; tt < 4; ++tt) {
            const int t20 = (wv * 4 + tt) * 16;
            v2f bb;
            bb.x = keb[(size_t)c * NT2 + t20 + r];
            bb.y = keb[(size_t)(c + 1) * NT2 + t20 + r];
            acc[tt] = WMMA_F32(a, bb, acc[tt]);
        }
    }
    for (int tt = 0; tt < 4; ++tt) {
        const int col = (wv * 4 + tt) * 16 + r;
        const float kk = k2[b * NT2 + col];
        for (int v = 0; v < 8; ++v) {
            const int m = v + 8 * h;
            const float qq = q2[b * NT1 + t10 + m];
            sc[m][col] = -TEMP * (qq + kk - 2.0f * acc[tt][v]);
        }
    }
    __syncthreads();

    // Phase 2: 16 threads per row, 32 columns each.
    const int row = tid >> 4;
    const int seg = tid & 15;
    const int t1  = t10 + row;
    float* srow = sc[row];
    const size_t obase = ((size_t)b * NT1 + t1) * NT2;

    float m1 = -3.0e38f;
    for (int j = 0; j < 32; ++j) m1 = fmaxf(m1, srow[seg * 32 + j]);
    red[row][seg] = m1;
    __syncthreads();
    float rm = red[row][0];
    for (int i = 1; i < 16; ++i) rm = fmaxf(rm, red[row][i]);
    __syncthreads();
    float ps = 0.0f;
    for (int j = 0; j < 32; ++j) ps += expf(srow[seg * 32 + j] - rm);
    red[row][seg] = ps;
    __syncthreads();
    float rs = 0.0f;
    for (int i = 0; i < 16; ++i) rs += red[row][i];
    const float lse = logf(rs);

    for (int j = 0; j < 32; ++j) {
        const int col = seg * 32 + j;
        const float lp = srow[col] - rm - lse + logf(prior[obase + col] + 1e-8f);
        logp_out[obase + col] = lp;
        srow[col] = mask[b * NT2 + col] ? -1e9f : lp;
    }
    __syncthreads();

    float m2 = -3.0e38f;
    for (int j = 0; j < 32; ++j) m2 = fmaxf(m2, srow[seg * 32 + j]);
    red[row][seg] = m2;
    __syncthreads();
    float rm2 = red[row][0];
    for (int i = 1; i < 16; ++i) rm2 = fmaxf(rm2, red[row][i]);
    __syncthreads();
    float ps2 = 0.0f;
    for (int j = 0; j < 32; ++j) ps2 += expf(srow[seg * 32 + j] - rm2);
    red[row][seg] = ps2;
    __syncthreads();
    float rs2 = 0.0f;
    for (int i = 0; i < 16; ++i) rs2 += red[row][i];
    const float inv = 1.0f / rs2;
    for (int j = 0; j < 32; ++j) {
        const int col = seg * 32 + j;
        attn_out[obase + col] = expf(srow[col] - rm2) * inv;
    }
}

// ---------------------------------------------------------------------------
// Launch. Input order follows setup_inputs():
// 0 queries 1 keys 2 mask 3 attn_prior 4 speaker_embed
// 5 kw1 6 kb1 7 kw2 8 kb2 9 qw1 10 qb1 11 qw2 12 qb2 13 qw3 14 qb3
// 15 kspk_w 16 kspk_b 17 qspk_w 18 qspk_b
// Output: attn [32,1,2048,512] then attn_logprob, concatenated.
// ---------------------------------------------------------------------------
extern "C" void kernel_launch(void* const* d_in, const int* in_sizes, int n_in,
                              void* d_out, int out_size, void* d_ws, size_t ws_size,
                              hipStream_t stream)
{
    const float* queries = (const float*)d_in[0];
    const float* keys    = (const float*)d_in[1];
    const unsigned char* mask = (const unsigned char*)d_in[2];
    const float* prior   = (const float*)d_in[3];
    const float* emb     = (const float*)d_in[4];
    const float* kw1 = (const float*)d_in[5];  const float* kb1 = (const float*)d_in[6];
    const float* kw2 = (const float*)d_in[7];  const float* kb2 = (const float*)d_in[8];
    const float* qw1 = (const float*)d_in[9];  const float* qb1 = (const float*)d_in[10];
    const float* qw2 = (const float*)d_in[11]; const float* qb2 = (const float*)d_in[12];
    const float* qw3 = (const float*)d_in[13]; const float* qb3 = (const float*)d_in[14];
    const float* kspk_w = (const float*)d_in[15]; const float* kspk_b = (const float*)d_in[16];
    const float* qspk_w = (const float*)d_in[17]; const float* qspk_b = (const float*)d_in[18];

    // Workspace layout (floats). Big conv3 hidden buffer reused by both paths
    // (stream-serial order makes this safe). Total ~23.4M floats (~94 MB).
    float* ws       = (float*)d_ws;
    float* kspk     = ws;                 // 32*512        = 16384
    float* qspk     = ws + 16384;         // 32*80         = 2560
    float* keys_enc = ws + 18944;         // 32*80*512     = 1310720
    float* qe       = ws + 1329664;       // 32*80*2048    = 5242880
    float* q2       = ws + 6572544;       // 32*2048       = 65536
    float* k2       = ws + 6638080;       // 32*512        = 16384
    float* big      = ws + 6654464;       // max 32*1024*512 = 16777216
    float* kh1 = big;                     // keys hidden  [32,1024,512]
    float* qh1 = big;                     // query hidden [32,160,2048] (reuse)
    float* qh2 = big + 10485760;          // query hidden [32,80,2048]

    float* attn_out = (float*)d_out;
    float* logp_out = attn_out + (size_t)NB * NT1 * NT2;

    // 1) speaker projections
    spk_proj<<<NB, 608, 0, stream>>>(emb, kspk_w, kspk_b, qspk_w, qspk_b, kspk, qspk);

    // 2) key path: conv3 512->1024 (+ReLU, +k_spk fused), conv1 1024->80
    conv3_relu_wmma<<<dim3(NT2 / 128, 1024 / 16, NB), 256, 0, stream>>>(
        keys, kspk, kw1, kb1, kh1, 512, 1024, NT2);
    gemm_conv1_wmma<<<dim3(NT2 / 128, NC / 16, NB), 256, 0, stream>>>(
        kh1, kw2, kb2, keys_enc, 1024, NC, NT2, 0);

    // 3) query path: conv3 80->160 (+ReLU, +q_spk), conv1 160->80 (+ReLU), conv1 80->80
    conv3_relu_wmma<<<dim3(NT1 / 128, 160 / 16, NB), 256, 0, stream>>>(
        queries, qspk, qw1, qb1, qh1, NC, 160, NT1);
    gemm_conv1_wmma<<<dim3(NT1 / 128, NC / 16, NB), 256, 0, stream>>>(
        qh1, qw2, qb2, qh2, 160, NC, NT1, 1);
    gemm_conv1_wmma<<<dim3(NT1 / 128, NC / 16, NB), 256, 0, stream>>>(
        qh2, qw3, qb3, qe, NC, NC, NT1, 0);

    // 4) squared norms
    sqsum<<<(NB * NT1) / 256, 256, 0, stream>>>(qe, q2, NC, NT1);
    sqsum<<<(NB * NT2) / 256, 256, 0, stream>>>(keys_enc, k2, NC, NT2);

    // 5) fused qk WMMA + double softmax + prior + mask
    attn_fused<<<dim3(NT1 / 16, NB), 256, 0, stream>>>(
        qe, keys_enc, q2, k2, prior, mask, attn_out, logp_out);
}